// QuantizedSparseLinear_6554120093747
// MI455X (gfx1250) — compile-verified
//
#include <hip/hip_runtime.h>

#define N_ROWS 4096
#define M_IN   4096
#define NNZ_T  1638400
#define B_SZ   128

typedef float v2f __attribute__((ext_vector_type(2)));
typedef float v8f __attribute__((ext_vector_type(8)));

// ---------------- Kernel 1: row_ptr[n] = lower_bound(row_ids, n) ----------------
__global__ void qsl_build_rowptr(const int* __restrict__ row_ids,
                                 int* __restrict__ row_ptr) {
    int n = blockIdx.x * blockDim.x + threadIdx.x;
    if (n > N_ROWS) return;
    int lo = 0, hi = NNZ_T;
    while (lo < hi) {
        int mid = (lo + hi) >> 1;
        if (row_ids[mid] < n) lo = mid + 1; else hi = mid;
    }
    row_ptr[n] = lo;
}

// ---------------- Kernel 2: xT[m*B + b] = x[b*M + m] (LDS-tiled) ----------------
__global__ void qsl_transpose_x(const float* __restrict__ x,
                                float* __restrict__ xT) {
    __shared__ float tile[32][33];
    const int m0 = blockIdx.x * 32;
    const int b0 = blockIdx.y * 32;
    #pragma unroll
    for (int j = 0; j < 32; j += 8)
        tile[threadIdx.y + j][threadIdx.x] =
            x[(size_t)(b0 + threadIdx.y + j) * M_IN + (m0 + threadIdx.x)];
    __syncthreads();
    #pragma unroll
    for (int j = 0; j < 32; j += 8)
        xT[(size_t)(m0 + threadIdx.y + j) * B_SZ + (b0 + threadIdx.x)] =
            tile[threadIdx.x][threadIdx.y + j];
}

// ---------------- Kernel 3: SpMM via V_WMMA_F32_16X16X4_F32 ----------------
// One block (8 waves) per output row n. Wave w owns batch tile b0 = 16*w.
// Chunks of 4 nnz at 16B-aligned bases; NNZ % 4 == 0 guarantees aligned int4
// loads of col/val are always in-bounds, so only the weight VALUES need
// masking (branch-free) on the partial head/tail chunks. Gathers use a
// uniform SGPR base + 32-bit VGPR offset (xT is only 2 MB).

// Fast path: fully interior chunk, no masking.
__device__ __forceinline__ v8f qsl_chunk_fast(const float* __restrict__ xT,
                                              const int* __restrict__ wval,
                                              const int* __restrict__ colidx,
                                              int kb, unsigned bl, bool hi, v8f acc) {
    const int4 c4 = *(const int4*)(colidx + kb);   // uniform -> s_load_b128
    const int4 w4 = *(const int4*)(wval   + kb);   // uniform -> s_load_b128
    const unsigned offA = (unsigned)(hi ? c4.z : c4.x) * (unsigned)B_SZ + bl;
    const unsigned offB = (unsigned)(hi ? c4.w : c4.y) * (unsigned)B_SZ + bl;
    v2f A;
    A.x = xT[offA];                                // saddr + 32-bit voffset
    A.y = xT[offB];
    v2f Bm;
    Bm.x = (float)(hi ? w4.y : w4.x);
    Bm.y = (float)(hi ? w4.w : w4.z);
    return __builtin_amdgcn_wmma_f32_16x16x4_f32(false, A, false, Bm,
                                                 (short)0, acc, false, false);
}

// Masked path: zero weight values whose global index falls outside [s, e).
__device__ __forceinline__ v8f qsl_chunk_masked(const float* __restrict__ xT,
                                                const int* __restrict__ wval,
                                                const int* __restrict__ colidx,
                                                int kb, int s, int e,
                                                unsigned bl, bool hi, v8f acc) {
    const int4 c4 = *(const int4*)(colidx + kb);
    const int4 w4 = *(const int4*)(wval   + kb);
    const float v0 = (kb + 0 >= s && kb + 0 < e) ? (float)w4.x : 0.f;
    const float v1 = (kb + 1 >= s && kb + 1 < e) ? (float)w4.y : 0.f;
    const float v2 = (kb + 2 >= s && kb + 2 < e) ? (float)w4.z : 0.f;
    const float v3 = (kb + 3 >= s && kb + 3 < e) ? (float)w4.w : 0.f;
    const unsigned offA = (unsigned)(hi ? c4.z : c4.x) * (unsigned)B_SZ + bl;
    const unsigned offB = (unsigned)(hi ? c4.w : c4.y) * (unsigned)B_SZ + bl;
    v2f A;
    A.x = xT[offA];
    A.y = xT[offB];
    v2f Bm;
    Bm.x = hi ? v1 : v0;
    Bm.y = hi ? v3 : v2;
    return __builtin_amdgcn_wmma_f32_16x16x4_f32(false, A, false, Bm,
                                                 (short)0, acc, false, false);
}

__global__ void qsl_spmm_wmma(const float* __restrict__ xT,
                              const int*   __restrict__ wval,
                              const float* __restrict__ scales,
                              const int*   __restrict__ colidx,
                              const float* __restrict__ bias,
                              const int*   __restrict__ row_ptr,
                              float* __restrict__ out) {
    const int n    = blockIdx.x;
    const int wave = threadIdx.x >> 5;
    const int lane = threadIdx.x & 31;
    const int b0   = wave * 16;
    const bool hi  = lane >= 16;
    const unsigned bl = (unsigned)(b0 + (lane & 15));

    const int s = row_ptr[n];
    const int e = row_ptr[n + 1];
    const float scale = scales[n];
    const float bv    = bias[n];

    v8f acc = {0.f, 0.f, 0.f, 0.f, 0.f, 0.f, 0.f, 0.f};

    if (s < e) {                         // uniform per block
        const int kb = s & ~3;           // head chunk base (16B aligned)
        const int ke = (e - 1) & ~3;     // tail chunk base
        if (kb == ke) {
            acc = qsl_chunk_masked(xT, wval, colidx, kb, s, e, bl, hi, acc);
        } else {
            acc = qsl_chunk_masked(xT, wval, colidx, kb, s, e, bl, hi, acc);
            #pragma unroll 2
            for (int k = kb + 4; k < ke; k += 4) {
                __builtin_prefetch(colidx + k + 64, 0, 3);
                __builtin_prefetch(wval   + k + 64, 0, 3);
                acc = qsl_chunk_fast(xT, wval, colidx, k, bl, hi, acc);
            }
            acc = qsl_chunk_masked(xT, wval, colidx, ke, s, e, bl, hi, acc);
        }
    }

    // D layout: lane 0 VGPR r = y[b0+r]; lane 16 VGPR r = y[b0+8+r] (all 16 N-cols equal).
    if (lane == 0 || lane == 16) {
        const int bb = b0 + (hi ? 8 : 0);
        out[(size_t)(bb + 0) * N_ROWS + n] = acc[0] * scale + bv;
        out[(size_t)(bb + 1) * N_ROWS + n] = acc[1] * scale + bv;
        out[(size_t)(bb + 2) * N_ROWS + n] = acc[2] * scale + bv;
        out[(size_t)(bb + 3) * N_ROWS + n] = acc[3] * scale + bv;
        out[(size_t)(bb + 4) * N_ROWS + n] = acc[4] * scale + bv;
        out[(size_t)(bb + 5) * N_ROWS + n] = acc[5] * scale + bv;
        out[(size_t)(bb + 6) * N_ROWS + n] = acc[6] * scale + bv;
        out[(size_t)(bb + 7) * N_ROWS + n] = acc[7] * scale + bv;
    }
}

extern "C" void kernel_launch(void* const* d_in, const int* in_sizes, int n_in,
                              void* d_out, int out_size, void* d_ws, size_t ws_size,
                              hipStream_t stream) {
    const float* x       = (const float*)d_in[0];   // [B, M]
    const int*   wval    = (const int*)  d_in[1];   // [NNZ]
    const float* scales  = (const float*)d_in[2];   // [N]
    const int*   row_ids = (const int*)  d_in[3];   // [NNZ] sorted
    const int*   colidx  = (const int*)  d_in[4];   // [NNZ]
    const float* bias    = (const float*)d_in[5];   // [N]
    float*       out     = (float*)d_out;           // [B, N]

    // Workspace: row_ptr (N+1 ints) then xT[M, B] floats (256B aligned).
    char* ws = (char*)d_ws;
    int*   row_ptr = (int*)ws;
    float* xT      = (float*)(ws + ((((N_ROWS + 1) * sizeof(int)) + 255) & ~size_t(255)));

    // 1) CSR row pointers via binary search over sorted row ids.
    qsl_build_rowptr<<<(N_ROWS + 1 + 255) / 256, 256, 0, stream>>>(row_ids, row_ptr);

    // 2) Transpose x so per-nnz batch gathers are coalesced.
    dim3 tb(32, 8);
    dim3 tg(M_IN / 32, B_SZ / 32);
    qsl_transpose_x<<<tg, tb, 0, stream>>>(x, xT);

    // 3) SpMM: one block of 8 waves per output row.
    qsl_spmm_wmma<<<N_ROWS, 256, 0, stream>>>(xT, wval, scales, colidx, bias, row_ptr, out);
}